// CorrelationBlock_70720931495965
// MI455X (gfx1250) — compile-verified
//
#include <hip/hip_runtime.h>

// ---------------------------------------------------------------------------
// CorrelationBlock for MI455X (gfx1250, wave32).
//
// ok : (32,32,32,32) f32   fk : (32,64,64,64) f32
// corr[hw, xyz] = sum_b fsum[b,hw] * ok[b,xyz]     (GEMM M=4096,N=32768,K=32)
// ak[i, k']     = sum_b fsum2[b,i] * osum[b,k']    (GEMM M=2048,N=1024, K=32)
//   where fsum2[b,i] = fsum[b,2i]+fsum[b,2i+1], osum[b,k'] = sum_x ok[b,x,k']
// (the reshape+sum(axis=2) in the reference collapses to exactly this)
// Output: d_out = corr (134217728 f32)  ++  ak_hat (2097152 f32)
//
// Store-bandwidth bound: 536 MB corr write -> ~23 us floor @ 23.3 TB/s.
// GEMMs use V_WMMA_F32_16X16X4_F32; corr is written with non-temporal stores.
// ---------------------------------------------------------------------------

typedef __attribute__((ext_vector_type(2))) float v2f;
typedef __attribute__((ext_vector_type(8))) float v8f;

#define CORR_ELEMS 134217728   // 4096 * 32768
#define AK_ELEMS   2097152     // 2048 * 1024

// ---------------- stage 1: reductions of the inputs ----------------

// fsum[b*4096 + hw] = sum_d fk[(b*4096+hw)*64 + d]   (131072 outputs)
__global__ __launch_bounds__(256) void fsum_kernel(const float* __restrict__ fk,
                                                   float* __restrict__ fsum) {
  const int o = blockIdx.x * 256 + threadIdx.x;
  const float4* p = (const float4*)(fk + (size_t)o * 64);
  float s = 0.f;
#pragma unroll
  for (int i = 0; i < 16; ++i) {
    float4 v = p[i];
    s += (v.x + v.y) + (v.z + v.w);
  }
  fsum[o] = s;
}

// osum[b*1024 + k'] = sum_x ok[b*32768 + x*1024 + k']   (32768 outputs)
__global__ __launch_bounds__(256) void osum_kernel(const float* __restrict__ ok,
                                                   float* __restrict__ osum) {
  const int o = blockIdx.x * 256 + threadIdx.x;
  const int b = o >> 10, kp = o & 1023;
  const float* p = ok + (size_t)b * 32768 + kp;
  float s = 0.f;
#pragma unroll
  for (int x = 0; x < 32; ++x) s += p[x * 1024];
  osum[o] = s;
}

// fsum2[b*2048 + i] = fsum[b*4096 + 2i] + fsum[b*4096 + 2i+1]   (65536 outputs)
__global__ __launch_bounds__(256) void fsum2_kernel(const float* __restrict__ fsum,
                                                    float* __restrict__ fsum2) {
  const int o = blockIdx.x * 256 + threadIdx.x;
  const int b = o >> 11, i = o & 2047;
  fsum2[o] = fsum[b * 4096 + 2 * i] + fsum[b * 4096 + 2 * i + 1];
}

// ---------------- stage 2: rank-32 outer-product GEMM via WMMA ----------------
//
// D(MxN) = A^T(MxK) * B(KxN), K = 32, via 8 steps of V_WMMA_F32_16X16X4_F32.
//   A stored K-major: A[k*lda + m]   B stored K-major: B[k*ldb + n]
// Block = 256 threads = 8 waves; wave w owns M-tile (blockIdx.y*8+w)*16 and a
// 16x64 slab of N (4 accumulators) at n0 = blockIdx.x*64.
//
// ISA layouts (7.12.2): A/B 16x4 f32 = 2 VGPRs: idx = lane&15, K split
// {0,2}(lanes0-15/16-31) in vgpr0 and {1,3} in vgpr1. C/D: vgpr v holds
// row v + 8*(lane>>4), col lane&15.
template <bool NTSTORE>
__global__ __launch_bounds__(256) void wmma_gemm_k32(const float* __restrict__ A,
                                                     const float* __restrict__ B,
                                                     float* __restrict__ D,
                                                     int lda, int ldb, int ldn) {
  const int lane  = threadIdx.x & 31;
  const int wave  = threadIdx.x >> 5;
  const int m0    = (blockIdx.y * 8 + wave) * 16;
  const int n0    = blockIdx.x * 64;
  const int idx16 = lane & 15;          // M (for A) / N (for B) within tile
  const int khalf = (lane >> 4) * 2;    // K contribution from lane half: {0,2}

  v8f c0 = {}, c1 = {}, c2 = {}, c3 = {};

#pragma unroll
  for (int ks = 0; ks < 32; ks += 4) {
    const int k0 = ks + khalf;
    v2f a;
    a.x = A[(k0    ) * lda + m0 + idx16];
    a.y = A[(k0 + 1) * lda + m0 + idx16];
    const float* b0p = B + (size_t)(k0    ) * ldb + n0 + idx16;
    const float* b1p = B + (size_t)(k0 + 1) * ldb + n0 + idx16;
    v2f b0; b0.x = b0p[0];  b0.y = b1p[0];
    v2f b1; b1.x = b0p[16]; b1.y = b1p[16];
    v2f b2; b2.x = b0p[32]; b2.y = b1p[32];
    v2f b3; b3.x = b0p[48]; b3.y = b1p[48];
    c0 = __builtin_amdgcn_wmma_f32_16x16x4_f32(false, a, false, b0, (short)0, c0, false, false);
    c1 = __builtin_amdgcn_wmma_f32_16x16x4_f32(false, a, false, b1, (short)0, c1, false, false);
    c2 = __builtin_amdgcn_wmma_f32_16x16x4_f32(false, a, false, b2, (short)0, c2, false, false);
    c3 = __builtin_amdgcn_wmma_f32_16x16x4_f32(false, a, false, b3, (short)0, c3, false, false);
  }

  const int rbase = m0 + (lane >> 4) * 8;
#pragma unroll
  for (int v = 0; v < 8; ++v) {
    float* drow = D + (size_t)(rbase + v) * ldn + n0 + idx16;
    if (NTSTORE) {
      __builtin_nontemporal_store(c0[v], drow);
      __builtin_nontemporal_store(c1[v], drow + 16);
      __builtin_nontemporal_store(c2[v], drow + 32);
      __builtin_nontemporal_store(c3[v], drow + 48);
    } else {
      drow[0]  = c0[v];
      drow[16] = c1[v];
      drow[32] = c2[v];
      drow[48] = c3[v];
    }
  }
}

// ---------------- stage 3: deterministic mean + mask + normalize ----------------

__global__ __launch_bounds__(256) void reduce_partial(const float* __restrict__ ak,
                                                      float* __restrict__ part) {
  __shared__ float sm[256];
  float s = 0.f;
  for (int i = blockIdx.x * 256 + threadIdx.x; i < AK_ELEMS; i += 256 * 256)
    s += ak[i];
  sm[threadIdx.x] = s;
  __syncthreads();
  for (int off = 128; off > 0; off >>= 1) {
    if ((int)threadIdx.x < off) sm[threadIdx.x] += sm[threadIdx.x + off];
    __syncthreads();
  }
  if (threadIdx.x == 0) part[blockIdx.x] = sm[0];
}

__global__ __launch_bounds__(256) void reduce_final(const float* __restrict__ part,
                                                    float* __restrict__ avg) {
  __shared__ float sm[256];
  sm[threadIdx.x] = part[threadIdx.x];
  __syncthreads();
  for (int off = 128; off > 0; off >>= 1) {
    if ((int)threadIdx.x < off) sm[threadIdx.x] += sm[threadIdx.x + off];
    __syncthreads();
  }
  if (threadIdx.x == 0) avg[0] = sm[0] * (1.0f / (float)AK_ELEMS);
}

// ak (32,64,1024): mask by global mean, then per-(b',k') min/max over h'.
// In-place: one thread owns one (b',k') column of 64 strided elements.
__global__ __launch_bounds__(256) void akhat_kernel(float* __restrict__ akio,
                                                    const float* __restrict__ avgp) {
  const int t = blockIdx.x * 256 + threadIdx.x;  // 32768 columns
  const int bp = t >> 10, kp = t & 1023;
  float* col = akio + (size_t)bp * 65536 + kp;   // stride 1024 per h'
  const float avg = avgp[0];

  float mn = 3.402823466e38f, mx = -3.402823466e38f;
  for (int h = 0; h < 64; ++h) {
    float v = col[h * 1024];
    float w = (v > avg) ? v : 0.0f;
    mn = fminf(mn, w);
    mx = fmaxf(mx, w);
  }
  const float scale = mx - mn;  // ref divides post-subtraction: max-min
  for (int h = 0; h < 64; ++h) {
    float v = col[h * 1024];
    float w = (v > avg) ? v : 0.0f;
    col[h * 1024] = (w - mn) / scale;
  }
}

// ---------------------------------------------------------------------------

extern "C" void kernel_launch(void* const* d_in, const int* in_sizes, int n_in,
                              void* d_out, int out_size, void* d_ws, size_t ws_size,
                              hipStream_t stream) {
  const float* ok = (const float*)d_in[0];  // 32*32*32*32
  const float* fk = (const float*)d_in[1];  // 32*64*64*64
  float* out = (float*)d_out;
  float* ws  = (float*)d_ws;

  // workspace layout (floats), total ~0.92 MB
  float* fsum  = ws;            // 131072
  float* fsum2 = ws + 131072;   // 65536
  float* osum  = ws + 196608;   // 32768
  float* part  = ws + 229376;   // 256
  float* avg   = ws + 229632;   // 1

  float* corr = out;                 // 134217728 floats
  float* ak   = out + CORR_ELEMS;    // 2097152 floats (ak -> ak_hat in place)

  fsum_kernel<<<512, 256, 0, stream>>>(fk, fsum);
  osum_kernel<<<128, 256, 0, stream>>>(ok, osum);
  fsum2_kernel<<<256, 256, 0, stream>>>(fsum, fsum2);

  // corr = fsum^T (4096x32) @ ok (32x32768); NT stores for the 536 MB stream
  wmma_gemm_k32<true><<<dim3(512, 32), 256, 0, stream>>>(fsum, ok, corr,
                                                         4096, 32768, 32768);
  // ak = fsum2^T (2048x32) @ osum (32x1024); re-read soon, keep cached
  wmma_gemm_k32<false><<<dim3(16, 16), 256, 0, stream>>>(fsum2, osum, ak,
                                                         2048, 1024, 1024);

  reduce_partial<<<256, 256, 0, stream>>>(ak, part);
  reduce_final<<<1, 256, 0, stream>>>(part, avg);
  akhat_kernel<<<128, 256, 0, stream>>>(ak, avg);
}